// EdgeDetector_57681410785560
// MI455X (gfx1250) — compile-verified
//
#include <hip/hip_runtime.h>
#include <cstdint>
#include <cstddef>

// Problem constants (fixed by the reference setup_inputs()).
constexpr int IMG_H = 2048;
constexpr int IMG_W = 2048;
constexpr int BATCH = 16;

constexpr int TILE = 64;        // output tile is TILE x TILE
constexpr int IT = TILE + 6;    // 70: image tile (3-halo each side)
constexpr int IS = IT + 1;      // 71: odd stride -> no LDS bank conflicts
constexpr int ST = TILE + 4;    // 68: smoothed tile (2-halo)
constexpr int SS = ST + 1;      // 69
constexpr int ET = TILE + 2;    // 66: edge-magnitude tile (1-halo)
constexpr int ES = ET + 1;      // 67

__global__ __launch_bounds__(256)
void edge_detect_fused(const float* __restrict__ img, float* __restrict__ out) {
    __shared__ float smem[IT * IS + ST * SS + ET * ES];
    float* s_img = smem;                 // 70 x 71
    float* s_sm  = smem + IT * IS;       // 68 x 69
    float* s_ed  = s_sm + ST * SS;       // 66 x 67

    const int tid = threadIdx.x;
    const int ox0 = blockIdx.x * TILE;
    const int oy0 = blockIdx.y * TILE;
    const int b   = blockIdx.z;
    const float* imgB = img + (size_t)b * IMG_H * IMG_W;

    // ---- Stage 0: async global -> LDS tile load (row-linear => coalesced DMA),
    //      zero-filled halo outside the image (conv 'SAME' padding of the image).
    const uint32_t ldsBase = (uint32_t)(uintptr_t)s_img;  // low 32 bits of flat addr = LDS offset
    for (int i = tid; i < IT * IT; i += 256) {
        int r = i / IT, c = i - r * IT;
        int gy = oy0 - 3 + r;
        int gx = ox0 - 3 + c;
        uint32_t lds = ldsBase + (uint32_t)(r * IS + c) * 4u;
        if ((unsigned)gy < (unsigned)IMG_H && (unsigned)gx < (unsigned)IMG_W) {
            unsigned long long gp =
                (unsigned long long)(uintptr_t)(imgB + (size_t)gy * IMG_W + gx);
            asm volatile("global_load_async_to_lds_b32 %0, %1, off"
                         :: "v"(lds), "v"(gp) : "memory");
        } else {
            s_img[r * IS + c] = 0.0f;
        }
    }
    asm volatile("s_wait_asynccnt 0x0" ::: "memory");
    __syncthreads();

    // ---- Stage 1: separable Gaussian (1/16 [1 2 1]^T [1 2 1]); vertical sweep,
    //      horizontal row sums kept in registers. 68 cols x 4 strips x 17 rows.
    for (int s = tid; s < ST * 4; s += 256) {
        int c  = s >> 2;                 // column in smoothed tile
        int r0 = (s & 3) * 17;           // row strip start
        int gx = ox0 - 2 + c;
        bool colOK = (unsigned)gx < (unsigned)IMG_W;
        const float* p = s_img + c;      // smoothed row r uses img rows r..r+2, cols c..c+2
        float h0 = p[r0 * IS]       + 2.0f * p[r0 * IS + 1]       + p[r0 * IS + 2];
        float h1 = p[(r0 + 1) * IS] + 2.0f * p[(r0 + 1) * IS + 1] + p[(r0 + 1) * IS + 2];
        for (int r = r0; r < r0 + 17; ++r) {
            const float* q = p + (r + 2) * IS;
            float h2 = q[0] + 2.0f * q[1] + q[2];
            int gy = oy0 - 2 + r;
            float v = 0.0f;              // smoothed == 0 outside the image
            if (colOK && (unsigned)gy < (unsigned)IMG_H)
                v = (h0 + 2.0f * h1 + h2) * 0.0625f;
            s_sm[r * SS + c] = v;
            h0 = h1; h1 = h2;
        }
    }
    __syncthreads();

    // ---- Stage 2: separable Sobel X/Y + magnitude; vertical sweep with two
    //      register chains. 66 cols x 4 strips (17,17,17,15 rows).
    for (int s = tid; s < ET * 4; s += 256) {
        int c    = s >> 2;
        int rs   = s & 3;
        int r0   = rs * 17;
        int rlen = (rs == 3) ? 15 : 17;
        int gx = ox0 - 1 + c;
        bool colOK = (unsigned)gx < (unsigned)IMG_W;
        const float* p = s_sm + c;       // edge row r uses sm rows r..r+2, cols c..c+2
        float q0 = p[r0 * SS], q1 = p[r0 * SS + 1], q2 = p[r0 * SS + 2];
        float ha0 = q2 - q0;                       // horiz [-1 0 1]
        float hb0 = q0 + 2.0f * q1 + q2;           // horiz [ 1 2 1]
        q0 = p[(r0 + 1) * SS]; q1 = p[(r0 + 1) * SS + 1]; q2 = p[(r0 + 1) * SS + 2];
        float ha1 = q2 - q0;
        float hb1 = q0 + 2.0f * q1 + q2;
        for (int r = r0; r < r0 + rlen; ++r) {
            const float* q = p + (r + 2) * SS;
            q0 = q[0]; q1 = q[1]; q2 = q[2];
            float ha2 = q2 - q0;
            float hb2 = q0 + 2.0f * q1 + q2;
            int gy = oy0 - 1 + r;
            float e = 0.0f;
            if (colOK && (unsigned)gy < (unsigned)IMG_H) {
                float gxv = ha0 + 2.0f * ha1 + ha2;  // SOBEL_X (cross-corr)
                float gyv = hb2 - hb0;               // SOBEL_Y (cross-corr)
                e = sqrtf(gxv * gxv + gyv * gyv);
            }
            s_ed[r * ES + c] = e;
            ha0 = ha1; ha1 = ha2; hb0 = hb1; hb1 = hb2;
        }
    }
    __syncthreads();

    // ---- Stage 3: 8-neighbor NMS, vertical sweep; c = tid&63 keeps global
    //      stores coalesced. 64 cols x 4 strips x 16 rows (one strip/thread).
    {
        int c  = tid & 63;
        int r0 = (tid >> 6) * 16;
        int gx = ox0 + c;
        const float* p = s_ed + c;       // output row r uses ed rows r..r+2, cols c..c+2
        // Per row k: f = max(q0,q1,q2); g = max(q0,q2) (center excluded); cen = q1.
        float q0 = p[r0 * ES], q1 = p[r0 * ES + 1], q2 = p[r0 * ES + 2];
        float f0 = fmaxf(fmaxf(q0, q1), q2);
        q0 = p[(r0 + 1) * ES]; q1 = p[(r0 + 1) * ES + 1]; q2 = p[(r0 + 1) * ES + 2];
        float f1   = fmaxf(fmaxf(q0, q1), q2);
        float g1   = fmaxf(q0, q2);
        float cen1 = q1;
        float* dst = out + ((size_t)b * IMG_H + (size_t)(oy0 + r0)) * IMG_W + gx;
        for (int r = r0; r < r0 + 16; ++r) {
            const float* q = p + (r + 2) * ES;
            q0 = q[0]; q1 = q[1]; q2 = q[2];
            float f2 = fmaxf(fmaxf(q0, q1), q2);
            float g2 = fmaxf(q0, q2);
            int gy = oy0 + r;
            float res;
            if (gy == 0 || gy == IMG_H - 1 || gx == 0 || gx == IMG_W - 1) {
                res = cen1;                           // border: edges unchanged
            } else {
                float m = fmaxf(fmaxf(f0, g1), f2);   // max of 8 neighbors
                res = (cen1 < m) ? 0.0f : cen1;
            }
            *dst = res;
            dst += IMG_W;
            f0 = f1; f1 = f2; g1 = g2; cen1 = q1;
        }
    }
}

extern "C" void kernel_launch(void* const* d_in, const int* in_sizes, int n_in,
                              void* d_out, int out_size, void* d_ws, size_t ws_size,
                              hipStream_t stream) {
    (void)in_sizes; (void)n_in; (void)out_size; (void)d_ws; (void)ws_size;
    const float* img = (const float*)d_in[0];
    float* out = (float*)d_out;
    dim3 grid(IMG_W / TILE, IMG_H / TILE, BATCH);
    edge_detect_fused<<<grid, 256, 0, stream>>>(img, out);
}